// TaskAugPolicy_48120813584702
// MI455X (gfx1250) — compile-verified
//
#include <hip/hip_runtime.h>
#include <stdint.h>

#define B_DIM 128
#define C_DIM 12
#define T_DIM 5000
#define TQ (T_DIM / 4)          // 1250 float4 chunks per row
#define NOPS 7
#define NSTAGES 2
#define TWO_PI 6.28318530717958647692f

// ---------------- Async global<->LDS path (gfx1250, ASYNCcnt) ----------------
typedef int v4i __attribute__((vector_size(4 * sizeof(int))));
typedef __attribute__((address_space(1))) v4i GV4;   // global v4i
typedef __attribute__((address_space(3))) v4i LV4;   // LDS v4i

#if defined(__has_builtin)
#if __has_builtin(__builtin_amdgcn_global_load_async_to_lds_b128)
#define HAVE_ASYNC_LD 1
#endif
#if __has_builtin(__builtin_amdgcn_global_store_async_from_lds_b128)
#define HAVE_ASYNC_ST 1
#endif
#if __has_builtin(__builtin_amdgcn_s_wait_asynccnt)
#define HAVE_ASYNC_WAIT 1
#endif
#endif

__device__ __forceinline__ void async_g2l_b128(const void* g, void* l) {
#if defined(HAVE_ASYNC_LD)
  __builtin_amdgcn_global_load_async_to_lds_b128((GV4*)g, (LV4*)l, 0, 0);
#else
  uint32_t loff = (uint32_t)(uintptr_t)(__attribute__((address_space(3))) char*)l;
  asm volatile("global_load_async_to_lds_b128 %0, %1, off"
               :: "v"(loff), "v"(g) : "memory");
#endif
}

__device__ __forceinline__ void async_l2g_b128(void* g, const void* l) {
#if defined(HAVE_ASYNC_ST)
  __builtin_amdgcn_global_store_async_from_lds_b128((GV4*)g, (LV4*)l, 0, 0);
#else
  uint32_t loff = (uint32_t)(uintptr_t)(__attribute__((address_space(3))) char*)l;
  asm volatile("global_store_async_from_lds_b128 %0, %1, off"
               :: "v"(g), "v"(loff) : "memory");
#endif
}

__device__ __forceinline__ void wait_async0() {
#if defined(HAVE_ASYNC_WAIT)
  __builtin_amdgcn_s_wait_asynccnt(0);
#else
  asm volatile("s_wait_asynccnt 0x0" ::: "memory");
#endif
}

// ---------------- Counter-based RNG (deterministic per call) ----------------
__device__ __forceinline__ uint32_t pcg(uint32_t v) {
  uint32_t s = v * 747796405u + 2891336453u;
  uint32_t w = ((s >> ((s >> 28u) + 4u)) ^ s) * 277803737u;
  return (w >> 22u) ^ w;
}
__device__ __forceinline__ uint32_t rng3(uint32_t a, uint32_t b, uint32_t c) {
  return pcg(a + pcg(b + pcg(c)));
}
__device__ __forceinline__ float u01(uint32_t h) {  // (0,1)
  return ((float)(h >> 8) + 0.5f) * (1.0f / 16777216.0f);
}

// ---------------- Kernel: fused two-stage augmentation, LDS-resident row ----
// One block per (b,c) row: async-load the 20KB row into LDS, run both stages
// entirely out of LDS (all time-axis gathers are row-local), async-store once.
// Global traffic = read x once + write out once = roofline minimum (~61 MB).
__global__ __launch_bounds__(256) void aug_main(const float* __restrict__ x,
                                                const float* __restrict__ ws,
                                                float* __restrict__ out) {
  __shared__ float bufA[T_DIM];
  __shared__ float bufB[T_DIM];
  const int bc  = blockIdx.x;          // (b*C + c)
  const int b   = bc / C_DIM;
  const int tid = threadIdx.x;

  // async bulk copy: global row -> LDS (ASYNCcnt path, no VGPR staging)
  const float* xrow = x + (size_t)bc * T_DIM;
#pragma unroll 5
  for (int q = tid; q < TQ; q += 256)
    async_g2l_b128((const void*)(xrow + 4 * q), (void*)(bufA + 4 * q));
  wait_async0();
  __syncthreads();

  const float stepN = 1.0f / (float)(T_DIM - 1);
  float* src = bufA;
  float* dst = bufB;
#pragma unroll
  for (int k = 0; k < NSTAGES; ++k) {
    const float* W = ws + k * NOPS;
    const float w0 = W[0], w1 = W[1], w2 = W[2], w3 = W[3];
    const float w4 = W[4], w5 = W[5], w6 = W[6];
    const float* P = ws + 16 + (size_t)(k * B_DIM + b) * 12;
    const float mag0 = P[0], scale1 = P[1];
    const int   mstart = (int)P[2], mend = (int)P[3];
    const float mag3 = P[4], freq = P[5], ph = P[6];
    const float amp4 = P[7], wf = P[8], wp = P[9];
    const int   shift = (int)P[10];
    const uint32_t seedN = 0xB5297A4Du ^ (0x68E31DA4u * (uint32_t)(k * 1536 + bc));

    for (int t = tid; t < T_DIM; t += 256) {
      const float xv   = src[t];
      const float tf   = (float)t;
      const float t2pi = TWO_PI * tf * stepN;

      // op0: gaussian noise (Box-Muller, one 32-bit hash -> two 16-bit uniforms)
      const uint32_t h  = rng3(seedN, (uint32_t)t, 0x2545F491u);
      const float u1 = ((float)(h & 0xFFFFu) + 0.5f) * (1.0f / 65536.0f);
      const float u2 = ((float)(h >> 16)     + 0.5f) * (1.0f / 65536.0f);
      const float nrm = __fsqrt_rn(-2.0f * __logf(u1)) * __cosf(TWO_PI * u2);
      const float y0 = fmaf(mag0, nrm, xv);
      // op1: magnitude scale
      const float y1 = xv * scale1;
      // op2: time mask
      const float y2 = (t >= mstart && t < mend) ? 0.0f : xv;
      // op3: baseline wander
      const float y3 = fmaf(mag3, __sinf(fmaf(freq, t2pi, ph)), xv);
      // op4: temporal warp (bilinear gather from LDS)
      const float base = fmaf(2.0f * stepN, tf, -1.0f);
      const float disp = amp4 * __sinf(fmaf(wf, t2pi, wp));
      const float px = (fminf(fmaxf(base + disp, -1.0f), 1.0f) + 1.0f) *
                       (0.5f * (float)(T_DIM - 1));
      int i0 = (int)floorf(px);
      i0 = min(max(i0, 0), T_DIM - 1);
      const int i1 = min(i0 + 1, T_DIM - 1);
      const float frac = px - (float)i0;
      const float xl = src[i0], xr = src[i1];
      const float y4 = fmaf(xr - xl, frac, xl);
      // op5: circular shift (gather from LDS)
      int rt = t - shift;
      if (rt < 0) rt += T_DIM;
      const float y5 = src[rt];
      // weighted mix (op6 is identity)
      float y = w6 * xv;
      y = fmaf(w0, y0, y);
      y = fmaf(w1, y1, y);
      y = fmaf(w2, y2, y);
      y = fmaf(w3, y3, y);
      y = fmaf(w4, y4, y);
      y = fmaf(w5, y5, y);
      dst[t] = y;
    }
    __syncthreads();
    float* tmp = src; src = dst; dst = tmp;
  }

  // async bulk copy: LDS result (bufA after 2 swaps) -> global
  float* orow = out + (size_t)bc * T_DIM;
#pragma unroll 5
  for (int q = tid; q < TQ; q += 256)
    async_l2g_b128((void*)(orow + 4 * q), (const void*)(src + 4 * q));
  wait_async0();
}

// ---------------- Kernel: stage weights + per-sample params ----------------
// ws layout (floats): [0..13]  w[stage][op]
//                     [16 + (k*128+b)*12 ..] 12 params per (stage, sample)
__global__ void aug_params(const float* __restrict__ logits,
                           const float* __restrict__ mag_neg,
                           const float* __restrict__ mag_pos,
                           const int*   __restrict__ labels,
                           float* __restrict__ ws) {
  const int k = blockIdx.x;   // stage
  const int b = threadIdx.x;  // sample

  if (b == 0) {  // gumbel-softmax weights (uniform over block)
    float z[NOPS];
    float m = -1e30f;
    for (int i = 0; i < NOPS; ++i) {
      float u = u01(rng3(0x9E3779B9u * (uint32_t)(k + 1), 0xA5A5A5A5u, (uint32_t)i));
      float g = -__logf(-__logf(u));
      z[i] = logits[k * NOPS + i] + g;   // TAU == 1
      m = fmaxf(m, z[i]);
    }
    float s = 0.0f;
    for (int i = 0; i < NOPS; ++i) { z[i] = __expf(z[i] - m); s += z[i]; }
    const float inv = 1.0f / s;
    for (int i = 0; i < NOPS; ++i) ws[k * NOPS + i] = z[i] * inv;
  }

  const float lab = (float)labels[b];
  float mag[NOPS];
  for (int i = 0; i < NOPS; ++i) {
    const float mn = mag_neg[k * NOPS + i];
    const float mp = mag_pos[k * NOPS + i];
    mag[i] = fabsf(fmaf(lab, mp - mn, mn));
  }
  float* P = ws + 16 + (size_t)(k * B_DIM + b) * 12;
  P[0] = mag[0];                         // gaussian magnitude
  P[1] = 1.0f + mag[1];                  // scale
  int ml = (int)floorf(mag[2] * (float)T_DIM);
  ml = min(max(ml, 0), T_DIM);
  const float um = u01(rng3(0x51ED270Bu, (uint32_t)k, (uint32_t)b));
  const int start = (int)floorf(um * (float)max(T_DIM - ml + 1, 1));
  P[2] = (float)start;                   // mask start
  P[3] = (float)(start + ml);            // mask end
  P[4] = mag[3];                         // wander magnitude
  P[5] = u01(rng3(0xC2B2AE35u, (uint32_t)k, (uint32_t)b)) * 0.3f + 0.05f;  // freq
  P[6] = u01(rng3(0x27D4EB2Fu, (uint32_t)k, (uint32_t)b)) * TWO_PI;        // phase
  P[7] = 0.1f * mag[4];                  // warp amplitude
  P[8] = u01(rng3(0x165667B1u, (uint32_t)k, (uint32_t)b)) * 2.0f + 1.0f;   // warp freq
  P[9] = u01(rng3(0x85EBCA77u, (uint32_t)k, (uint32_t)b)) * TWO_PI;        // warp phase
  P[10] = floorf(mag[5] * (float)T_DIM); // circular shift
  P[11] = 0.0f;
}

extern "C" void kernel_launch(void* const* d_in, const int* in_sizes, int n_in,
                              void* d_out, int out_size, void* d_ws, size_t ws_size,
                              hipStream_t stream) {
  (void)in_sizes; (void)n_in; (void)out_size; (void)ws_size;
  const float* x       = (const float*)d_in[0];
  const float* logits  = (const float*)d_in[1];
  const float* mag_neg = (const float*)d_in[2];
  const float* mag_pos = (const float*)d_in[3];
  const int*   labels  = (const int*)d_in[4];
  float* out = (float*)d_out;
  float* ws  = (float*)d_ws;

  aug_params<<<dim3(NSTAGES), dim3(B_DIM), 0, stream>>>(logits, mag_neg, mag_pos,
                                                        labels, ws);
  aug_main<<<dim3(B_DIM * C_DIM), dim3(256), 0, stream>>>(x, ws, out);
}